// GCNWithJK_HIGH_58437325029847
// MI455X (gfx1250) — compile-verified
//
#include <hip/hip_runtime.h>
#include <hip/hip_bf16.h>
#include <math.h>

typedef float v2f __attribute__((ext_vector_type(2)));
typedef float v8f __attribute__((ext_vector_type(8)));

#define N_NODES 50000
#define N_EDGES 800000
#define HDIM 128
#define NGRAPH 512
#define NCLASS 10

// ---------------------------------------------------------------------------
// Fuse trained + "random" conv weights: linearity of GCN conv lets us replace
// 0.95*conv(w) + 0.05*conv(wr) with conv(0.95*w + 0.05*wr).
// ---------------------------------------------------------------------------
__global__ void prep_w_kernel(const float* __restrict__ w1, const float* __restrict__ wr1,
                              const float* __restrict__ w2, const float* __restrict__ w3,
                              const float* __restrict__ wr,
                              const float* __restrict__ b1, const float* __restrict__ br1,
                              const float* __restrict__ b2, const float* __restrict__ b3,
                              const float* __restrict__ br,
                              float* __restrict__ weff, float* __restrict__ beff) {
  int t = blockIdx.x * blockDim.x + threadIdx.x;
  if (t < HDIM * HDIM) {
    weff[t]               = 0.95f * w1[t] + 0.05f * wr1[t];
    weff[HDIM*HDIM + t]   = 0.95f * w2[t] + 0.05f * wr[t];
    weff[2*HDIM*HDIM + t] = 0.95f * w3[t] + 0.05f * wr[t];
  }
  if (t < HDIM) {
    beff[t]          = 0.95f * b1[t] + 0.05f * br1[t];
    beff[HDIM + t]   = 0.95f * b2[t] + 0.05f * br[t];
    beff[2*HDIM + t] = 0.95f * b3[t] + 0.05f * br[t];
  }
}

// deg starts at 1.0 (self-loop); zero pooled sums and per-graph counts.
__global__ void init_kernel(float* __restrict__ deg, float* __restrict__ pooled,
                            float* __restrict__ cnt, int n) {
  int t = blockIdx.x * blockDim.x + threadIdx.x;
  if (t < n)               deg[t] = 1.0f;
  if (t < NGRAPH * 3*HDIM) pooled[t] = 0.0f;
  if (t < NGRAPH)          cnt[t] = 0.0f;
}

__global__ void deg_kernel(const int* __restrict__ ei, float* __restrict__ deg, int E) {
  int e = blockIdx.x * blockDim.x + threadIdx.x;
  if (e < E) atomicAdd(&deg[ei[e]], 1.0f);
}

// dinv = rsqrt(deg) in place (deg >= 1 always, so the deg>0 guard is moot);
// also accumulate per-graph node counts for mean pooling.
__global__ void dinv_cnt_kernel(float* __restrict__ deg, const int* __restrict__ batch,
                                float* __restrict__ cnt, int n) {
  int i = blockIdx.x * blockDim.x + threadIdx.x;
  if (i >= n) return;
  deg[i] = rsqrtf(deg[i]);
  atomicAdd(&cnt[batch[i]], 1.0f);
}

// ---------------------------------------------------------------------------
// WMMA GEMM: Y = dinv[:,None] * (H @ W), ACC = Y (self-loop pre-accumulated).
// fp32 path: V_WMMA_F32_16X16X4_F32, one wave per 16x16 tile, 32 steps over K.
// A 16x4 layout : lane L holds row M=L%16, K = k0 + 2*(L/16) + {0,1}
// B 4x16 layout : lane L holds col N=L%16, K = k0 + 2*(L/16) + {0,1}
// C 16x16 layout: VGPR v, lane L -> M = v + 8*(L/16), N = L%16
// ---------------------------------------------------------------------------
__global__ void gemm_scale_kernel(const float* __restrict__ Hin, const float* __restrict__ W,
                                  const float* __restrict__ dinv, float* __restrict__ Y,
                                  float* __restrict__ ACC) {
  const int wave = threadIdx.x >> 5;     // 0..7 -> N tile
  const int lane = threadIdx.x & 31;
  const int half = lane >> 4;            // 0 or 1
  const int l16  = lane & 15;
  const int m0 = blockIdx.x * 16;
  const int n0 = wave * 16;

  const float* arow = Hin + (size_t)(m0 + l16) * HDIM;
  v8f c = {};
#pragma unroll
  for (int k0 = 0; k0 < HDIM; k0 += 4) {
    const int k = k0 + 2 * half;
    v2f a; a.x = arow[k];                     a.y = arow[k + 1];
    v2f b; b.x = W[(size_t)k * HDIM + n0 + l16];
           b.y = W[(size_t)(k + 1) * HDIM + n0 + l16];
    c = __builtin_amdgcn_wmma_f32_16x16x4_f32(false, a, false, b, (short)0, c,
                                              false, false);
  }
#pragma unroll
  for (int v = 0; v < 8; ++v) {
    const int mm = m0 + v + 8 * half;
    const float val = dinv[mm] * c[v];
    const size_t idx = (size_t)mm * HDIM + n0 + l16;
    Y[idx] = val;
    ACC[idx] = val;
  }
}

// One wave per edge: gather Y[col] (float4 per lane) and scatter-add into
// ACC[row] with f32 atomics. Whole node table lives in L2 (25.6 MB << 192 MB).
__global__ void edge_scatter_kernel(const int* __restrict__ ei, const float* __restrict__ Y,
                                    float* __restrict__ ACC, int E) {
  const int lane = threadIdx.x & 31;
  const int e = blockIdx.x * (blockDim.x >> 5) + (threadIdx.x >> 5);
  if (e >= E) return;
  const int r = ei[e];
  const int cidx = ei[E + e];
  const float4 v = *(const float4*)(Y + (size_t)cidx * HDIM + lane * 4);
  float* dst = ACC + (size_t)r * HDIM + lane * 4;
  atomicAdd(dst + 0, v.x);
  atomicAdd(dst + 1, v.y);
  atomicAdd(dst + 2, v.z);
  atomicAdd(dst + 3, v.w);
}

// h = relu(dinv*acc + b); store layer output and fuse the JK mean-pool
// accumulation (pooled[g, layer*128 + j] += h).
__global__ void finalize_kernel(const float* __restrict__ ACC, const float* __restrict__ dinv,
                                const float* __restrict__ b, const int* __restrict__ batch,
                                float* __restrict__ pooled, float* __restrict__ hout,
                                int n, int layer) {
  const int t = blockIdx.x * blockDim.x + threadIdx.x;
  if (t >= n * HDIM) return;
  const int i = t >> 7;
  const int j = t & (HDIM - 1);
  float val = dinv[i] * ACC[t] + b[j];
  val = val > 0.0f ? val : 0.0f;
  hout[t] = val;
  atomicAdd(&pooled[(size_t)batch[i] * (3*HDIM) + layer * HDIM + j], val);
}

// g = relu((pooled/cnt) @ lin1_w + lin1_b)   [512,384]x[384,128]
__global__ void lin1_kernel(const float* __restrict__ pooled, const float* __restrict__ cnt,
                            const float* __restrict__ w, const float* __restrict__ b,
                            float* __restrict__ g) {
  const int t = blockIdx.x * blockDim.x + threadIdx.x;
  if (t >= NGRAPH * HDIM) return;
  const int gr = t >> 7;
  const int j = t & (HDIM - 1);
  const float inv = 1.0f / fmaxf(cnt[gr], 1.0f);
  const float* p = pooled + (size_t)gr * (3*HDIM);
  float s = b[j];
  for (int k = 0; k < 3*HDIM; ++k) s = fmaf(p[k] * inv, w[(size_t)k * HDIM + j], s);
  g[t] = fmaxf(s, 0.0f);
}

// logits = g @ lin2_w + lin2_b; out = log_softmax(logits)   [512,128]x[128,10]
__global__ void lin2_lsm_kernel(const float* __restrict__ g, const float* __restrict__ w,
                                const float* __restrict__ b, float* __restrict__ out) {
  const int gr = blockIdx.x * blockDim.x + threadIdx.x;
  if (gr >= NGRAPH) return;
  const float* gv = g + (size_t)gr * HDIM;
  float logits[NCLASS];
#pragma unroll
  for (int c = 0; c < NCLASS; ++c) {
    float s = b[c];
    for (int k = 0; k < HDIM; ++k) s = fmaf(gv[k], w[(size_t)k * NCLASS + c], s);
    logits[c] = s;
  }
  float m = logits[0];
#pragma unroll
  for (int c = 1; c < NCLASS; ++c) m = fmaxf(m, logits[c]);
  float se = 0.0f;
#pragma unroll
  for (int c = 0; c < NCLASS; ++c) se += expf(logits[c] - m);
  const float lse = logf(se);
#pragma unroll
  for (int c = 0; c < NCLASS; ++c) out[(size_t)gr * NCLASS + c] = logits[c] - m - lse;
}

extern "C" void kernel_launch(void* const* d_in, const int* in_sizes, int n_in,
                              void* d_out, int out_size, void* d_ws, size_t ws_size,
                              hipStream_t stream) {
  const float* x      = (const float*)d_in[0];
  const int*   ei     = (const int*)d_in[1];     // int32 (JAX default), [2, E] flat
  const int*   batch  = (const int*)d_in[2];
  const float* w1     = (const float*)d_in[3];
  const float* b1     = (const float*)d_in[4];
  const float* wr1    = (const float*)d_in[5];
  const float* br1    = (const float*)d_in[6];
  const float* w2     = (const float*)d_in[7];
  const float* b2     = (const float*)d_in[8];
  const float* w3     = (const float*)d_in[9];
  const float* b3     = (const float*)d_in[10];
  const float* wr     = (const float*)d_in[11];
  const float* br     = (const float*)d_in[12];
  const float* lin1_w = (const float*)d_in[13];
  const float* lin1_b = (const float*)d_in[14];
  const float* lin2_w = (const float*)d_in[15];
  const float* lin2_b = (const float*)d_in[16];
  float* out = (float*)d_out;

  // Workspace carve-up (256 B aligned slices).
  char* p = (char*)d_ws;
  auto carve = [&](size_t bytes) -> float* {
    float* r = (float*)p;
    p += (bytes + 255) & ~(size_t)255;
    return r;
  };
  float* weff   = carve(3 * HDIM * HDIM * sizeof(float));
  float* beff   = carve(3 * HDIM * sizeof(float));
  float* dinv   = carve(N_NODES * sizeof(float));            // deg -> dinv in place
  float* cnt    = carve(NGRAPH * sizeof(float));
  float* pooled = carve((size_t)NGRAPH * 3*HDIM * sizeof(float));
  float* gbuf   = carve((size_t)NGRAPH * HDIM * sizeof(float));
  float* Y      = carve((size_t)N_NODES * HDIM * sizeof(float));
  float* ACC    = carve((size_t)N_NODES * HDIM * sizeof(float));
  float* hbuf   = carve((size_t)N_NODES * HDIM * sizeof(float));

  prep_w_kernel<<<(HDIM*HDIM + 255) / 256, 256, 0, stream>>>(w1, wr1, w2, w3, wr,
                                                             b1, br1, b2, b3, br, weff, beff);
  init_kernel<<<(NGRAPH*3*HDIM + 255) / 256, 256, 0, stream>>>(dinv, pooled, cnt, N_NODES);
  deg_kernel<<<(N_EDGES + 255) / 256, 256, 0, stream>>>(ei, dinv, N_EDGES);
  dinv_cnt_kernel<<<(N_NODES + 255) / 256, 256, 0, stream>>>(dinv, batch, cnt, N_NODES);

  const float* hin = x;
  for (int layer = 0; layer < 3; ++layer) {
    gemm_scale_kernel<<<N_NODES / 16, 256, 0, stream>>>(hin, weff + layer * HDIM * HDIM,
                                                        dinv, Y, ACC);
    edge_scatter_kernel<<<N_EDGES / 8, 256, 0, stream>>>(ei, Y, ACC, N_EDGES);
    // hbuf in-place across layers is safe: the GEMM that read hbuf has
    // completed (stream order) before finalize overwrites it.
    finalize_kernel<<<(N_NODES * HDIM + 255) / 256, 256, 0, stream>>>(
        ACC, dinv, beff + layer * HDIM, batch, pooled, hbuf, N_NODES, layer);
    hin = hbuf;
  }

  lin1_kernel<<<(NGRAPH * HDIM + 255) / 256, 256, 0, stream>>>(pooled, cnt, lin1_w, lin1_b, gbuf);
  lin2_lsm_kernel<<<(NGRAPH + 255) / 256, 256, 0, stream>>>(gbuf, lin2_w, lin2_b, out);
}